// BlockAllocator_77068893160287
// MI455X (gfx1250) — compile-verified
//
#include <hip/hip_runtime.h>

#define NBK 4
#define SEQ 32
#define BS 8
#define H 64
#define MSLOTS 6
#define BPW 8            // batch elements per workgroup
#define THREADS 256

typedef __attribute__((ext_vector_type(2))) float v2f;
typedef __attribute__((ext_vector_type(8))) float v8f;

static __device__ __forceinline__ v8f wmma4(v2f a, v2f b, v8f c) {
  // fp32 WMMA: D(16x16,f32) = A(16x4,f32) x B(4x16,f32) + C
  // (neg_a, A, neg_b, B, c_mod, C, reuse_a, reuse_b)
  return __builtin_amdgcn_wmma_f32_16x16x4_f32(false, a, false, b, (short)0, c, false, false);
}

// A fragment (16x4 f32) from row-major f32 source S[m][k] (ld floats/row).
// ISA 7.12.2 32-bit A 16x4: lanes 0-15 -> M=lane, v0=K0,v1=K1; lanes 16-31 -> M=lane-16, v0=K2,v1=K3.
static __device__ __forceinline__ v2f build_a4(const float* S, int ld, int row0,
                                               int k0, int lane, int mMax) {
  int m = row0 + (lane & 15);
  if (m > mMax) m = mMax;               // padded rows: clamp (results discarded)
  const int k = k0 + ((lane >> 4) << 1);
  const float2 t = *(const float2*)(S + m * ld + k);   // 8B-aligned (k even, ld even)
  v2f a; a[0] = t.x; a[1] = t.y;
  return a;
}

// B fragment (4x16 f32) from row-major f32 weight W[k][n] (ldn floats/row),
// columns >= ncols zero-padded. Same K split: lanes 0-15 -> K=k0,k0+1; lanes 16-31 -> K=k0+2,k0+3.
static __device__ __forceinline__ v2f build_b4(const float* W, int ldn, int ncols,
                                               int k0, int n0, int lane) {
  const int n = n0 + (lane & 15);
  const bool ok = n < ncols;
  const int nc = ok ? n : 0;
  const int k = k0 + ((lane >> 4) << 1);
  v2f b;
  b[0] = W[k * ldn + nc];
  b[1] = W[(k + 1) * ldn + nc];
  if (!ok) { b[0] = 0.f; b[1] = 0.f; }
  return b;
}

static __device__ __forceinline__ int sel4(int a0, int a1, int a2, int a3, int i) {
  return i == 0 ? a0 : i == 1 ? a1 : i == 2 ? a2 : a3;
}

__global__ void __launch_bounds__(THREADS)
block_allocator_kernel(const float* __restrict__ hidden,
                       const float* __restrict__ w_pool, const float* __restrict__ b_pool,
                       const float* __restrict__ wa1,    const float* __restrict__ ba1,
                       const float* __restrict__ wa2,    const float* __restrict__ ba2,
                       const float* __restrict__ ws1,    const float* __restrict__ bs1,
                       const float* __restrict__ ws2,    const float* __restrict__ bs2,
                       float* __restrict__ outMem, float* __restrict__ outMask,
                       float* __restrict__ outAlloc, int Btot)
{
  extern __shared__ float smem[];
  float* hid     = smem;                         // BPW*SEQ*H = 16384 f32 (fp32 kept for exact gather)
  float* pooledR = hid + BPW * SEQ * H;          // 32*64 = 2048
  float* pooledP = pooledR + NBK * BPW * H;      // 32*64 = 2048
  float* h1s     = pooledP + NBK * BPW * H;      // 16*64 = 1024
  float* logitsS = h1s + 16 * H;                 // 8*4 (pad 32)
  float* scoresS = logitsS + 32;                 // 256
  int*   selIdx  = (int*)(scoresS + BPW * SEQ);  // 48

  const int tid  = threadIdx.x;
  const int wave = tid >> 5;
  const int lane = tid & 31;
  const int hl   = lane >> 4;

  // ---- phase 0: stage hidden (fp32) into LDS, init scores with bs2 ----
  {
    for (int i = tid; i < BPW * SEQ * H / 4; i += THREADS) {
      int el = i >> 9;                                    // 512 float4 per element
      size_t eg = (size_t)blockIdx.x * BPW + el;
      if (eg >= (size_t)Btot) eg = (size_t)Btot - 1;      // tail clamp (dup loads ok)
      const float4* src = (const float4*)(hidden + eg * (size_t)(SEQ * H));
      ((float4*)hid)[i] = src[i & 511];
    }
    float b2 = bs2[0];
    for (int i = tid; i < BPW * SEQ; i += THREADS) scoresS[i] = b2;
  }
  __syncthreads();

  // ---- phase 0b: per-block mean pooling -> pooledR[e*4+nb][h] ----
  for (int i = tid; i < NBK * BPW * H; i += THREADS) {
    int row = i >> 6, col = i & 63;
    int e = row >> 2, nb = row & 3;
    const float* p = hid + ((e * SEQ) + nb * BS) * H + col;
    float s = 0.f;
#pragma unroll
    for (int t = 0; t < BS; ++t) s += p[t * H];
    pooledR[i] = s * (1.0f / BS);
  }
  __syncthreads();

  // ---- phase 1: pooledP = pooledR @ w_pool + b_pool  (M=32,K=64,N=64; 8 tiles / 8 waves) ----
  {
    int mt = wave >> 2, nt = wave & 3;
    v8f acc = {};
#pragma unroll
    for (int ks = 0; ks < 16; ++ks) {
      v2f a = build_a4(pooledR, H, mt * 16, ks * 4, lane, NBK * BPW - 1);
      v2f b = build_b4(w_pool, H, H, ks * 4, nt * 16, lane);
      acc = wmma4(a, b, acc);
    }
    int col = nt * 16 + (lane & 15);
    float bp = b_pool[col];
#pragma unroll
    for (int j = 0; j < 8; ++j) {
      int row = mt * 16 + 8 * hl + j;
      pooledP[row * H + col] = acc[j] + bp;
    }
  }
  __syncthreads();

  // ---- phase 2: h1 = relu(block_feats @ wa1 + ba1)  (M=8 pad16, K=256, N=64; waves 0-3) ----
  // block_feats[e][k] == pooledP viewed as (8, 256) row-major.
  if (wave < 4) {
    int nt = wave;
    v8f acc = {};
#pragma unroll
    for (int ks = 0; ks < 64; ++ks) {
      v2f a = build_a4(pooledP, NBK * H, 0, ks * 4, lane, BPW - 1);
      v2f b = build_b4(wa1, H, H, ks * 4, nt * 16, lane);
      acc = wmma4(a, b, acc);
    }
    int col = nt * 16 + (lane & 15);
    float bb = ba1[col];
#pragma unroll
    for (int j = 0; j < 8; ++j) {
      int row = 8 * hl + j;                  // rows 8..15 are pad (finite garbage)
      float v = acc[j] + bb;
      h1s[row * H + col] = v > 0.f ? v : 0.f;
    }
  }
  __syncthreads();

  // ---- phase 3: logits = h1 @ wa2 + ba2  (M=16, K=64, N=4 pad16; wave 0) ----
  if (wave == 0) {
    v8f acc = {};
#pragma unroll
    for (int ks = 0; ks < 16; ++ks) {
      v2f a = build_a4(h1s, H, 0, ks * 4, lane, 15);
      v2f b = build_b4(wa2, NBK, NBK, ks * 4, 0, lane);
      acc = wmma4(a, b, acc);
    }
    int n = lane & 15;
    if (n < NBK) {
      float bb = ba2[n];
#pragma unroll
      for (int j = 0; j < 8; ++j) {
        int m = 8 * hl + j;
        if (m < BPW) logitsS[m * NBK + n] = acc[j] + bb;
      }
    }
  }
  __syncthreads();

  // ---- phase 4: token scores = relu(hid @ ws1 + bs1) @ ws2  (M=256,K=64,N=32; 32 tiles) ----
  // nt = wave&1 is rep-invariant -> hoist the 16 ws1 B fragments out of the M loop.
  {
    const int nt = wave & 1;
    v2f bf[16];
#pragma unroll
    for (int ks = 0; ks < 16; ++ks)
      bf[ks] = build_b4(ws1, 32, 32, ks * 4, nt * 16, lane);

    const int col = nt * 16 + (lane & 15);
    const float w2 = ws2[col];
    const float b1 = bs1[col];

#pragma unroll
    for (int rep = 0; rep < 4; ++rep) {
      int mt = (wave >> 1) + 4 * rep;            // 4 waves per nt parity cover mt=0..15
      v8f acc = {};
#pragma unroll
      for (int ks = 0; ks < 16; ++ks) {
        v2f a = build_a4(hid, H, mt * 16, ks * 4, lane, BPW * SEQ - 1);
        acc = wmma4(a, bf[ks], acc);
      }
#pragma unroll
      for (int j = 0; j < 8; ++j) {
        int row = mt * 16 + 8 * hl + j;
        float t = acc[j] + b1;
        t = t > 0.f ? t : 0.f;
        atomicAdd(&scoresS[row], t * w2);        // ds_add_f32 (fused 2nd scorer layer)
      }
    }
  }
  __syncthreads();

  // ---- phase 5: per-element discretization + per-block top-k pack (wave == element) ----
  const int e = wave;
  const size_t eg = (size_t)blockIdx.x * BPW + e;

  float lg[4];
#pragma unroll
  for (int i = 0; i < 4; ++i) lg[i] = logitsS[e * NBK + i];
  float mx = fmaxf(fmaxf(lg[0], lg[1]), fmaxf(lg[2], lg[3]));
  float ex[4], ssum = 0.f;
#pragma unroll
  for (int i = 0; i < 4; ++i) { ex[i] = expf(lg[i] - mx); ssum += ex[i]; }
  float fl[4], fr[4]; float flsum = 0.f;
#pragma unroll
  for (int i = 0; i < 4; ++i) {
    float so = ex[i] / ssum * (float)MSLOTS;
    fl[i] = floorf(so); fr[i] = so - fl[i]; flsum += fl[i];
  }
  int rem = MSLOTS - (int)(flsum + 0.5f);
  if (rem < 0) rem = 0;
  int ia[4];
#pragma unroll
  for (int i = 0; i < 4; ++i) {
    int r = 0;
#pragma unroll
    for (int j = 0; j < 4; ++j)
      r += ((fr[j] > fr[i]) || (fr[j] == fr[i] && j < i)) ? 1 : 0;
    ia[i] = (int)fl[i] + (r < rem ? 1 : 0);
  }
  int pre1 = ia[0], pre2 = ia[0] + ia[1], pre3 = pre2 + ia[2];
  int cnt = pre3 + ia[3]; if (cnt > MSLOTS) cnt = MSLOTS;

  // lane = token; stable descending rank within its block of 8
  {
    int nb = lane >> 3, lt = lane & 7;
    float s = scoresS[e * SEQ + lane];
    int r = 0;
#pragma unroll
    for (int u = 0; u < BS; ++u) {
      float su = scoresS[e * SEQ + nb * BS + u];
      r += ((su > s) || (su == s && u < lt)) ? 1 : 0;
    }
    int ial = sel4(ia[0], ia[1], ia[2], ia[3], nb);
    int p   = sel4(0, pre1, pre2, pre3, nb) + r;
    if (r < ial && p < MSLOTS) selIdx[e * MSLOTS + p] = lane;
  }
  __syncthreads();
  if (lane == 0) {                            // pad-with-last / empty -> iota
    if (cnt == 0) {
      for (int j = 0; j < MSLOTS; ++j) selIdx[e * MSLOTS + j] = j;
    } else {
      int last = selIdx[e * MSLOTS + cnt - 1];
      for (int j = cnt; j < MSLOTS; ++j) selIdx[e * MSLOTS + j] = last;
    }
  }
  __syncthreads();

  // ---- phase 6: gather fp32 rows + mask + alloc ----
  if (eg < (size_t)Btot) {
    float* om = outMem + eg * (size_t)(MSLOTS * H);
#pragma unroll
    for (int j = 0; j < MSLOTS; ++j) {
      int idx = selIdx[e * MSLOTS + j];
      om[j * H + lane]      = hid[(e * SEQ + idx) * H + lane];
      om[j * H + lane + 32] = hid[(e * SEQ + idx) * H + lane + 32];
    }
    if (lane < MSLOTS) outMask[eg * MSLOTS + lane] = 1.0f;
    if (lane < NBK)    outAlloc[eg * NBK + lane] = (float)sel4(ia[0], ia[1], ia[2], ia[3], lane);
  }
}

extern "C" void kernel_launch(void* const* d_in, const int* in_sizes, int n_in,
                              void* d_out, int out_size, void* d_ws, size_t ws_size,
                              hipStream_t stream) {
  (void)n_in; (void)out_size; (void)d_ws; (void)ws_size;
  const float* hidden = (const float*)d_in[0];
  const float* w_pool = (const float*)d_in[1];
  const float* b_pool = (const float*)d_in[2];
  const float* wa1    = (const float*)d_in[3];
  const float* ba1    = (const float*)d_in[4];
  const float* wa2    = (const float*)d_in[5];
  const float* ba2    = (const float*)d_in[6];
  const float* ws1    = (const float*)d_in[7];
  const float* bs1    = (const float*)d_in[8];
  const float* ws2    = (const float*)d_in[9];
  const float* bs2    = (const float*)d_in[10];

  const int B = in_sizes[0] / (SEQ * H);
  if (B <= 0) return;

  float* out      = (float*)d_out;
  float* outMem   = out;
  float* outMask  = out + (size_t)B * MSLOTS * H;
  float* outAlloc = outMask + (size_t)B * MSLOTS;

  const int nwg = (B + BPW - 1) / BPW;
  const size_t shmem =
      (size_t)(BPW * SEQ * H + 2 * NBK * BPW * H + 16 * H + 32 + BPW * SEQ) * sizeof(float)
      + (size_t)(BPW * MSLOTS) * sizeof(int);

  block_allocator_kernel<<<nwg, THREADS, shmem, stream>>>(
      hidden, w_pool, b_pool, wa1, ba1, wa2, ba2, ws1, bs1, ws2, bs2,
      outMem, outMask, outAlloc, B);
}